// CLIR_31095563223418
// MI455X (gfx1250) — compile-verified
//
#include <hip/hip_runtime.h>

#define NUM_CLASSES 10
#define PAD_CLASSES 16   // S1 padded to 16 rows so WMMA feeds need no guards
#define DFEAT 128

typedef __attribute__((ext_vector_type(2))) float v2f;
typedef __attribute__((ext_vector_type(4))) float v4f;
typedef __attribute__((ext_vector_type(8))) float v8f;

// ---------------------------------------------------------------------------
// Kernel 1: zero the accumulator region of d_ws (harness does not re-zero).
// Rows 10..15 of S1 are zeroed here and never written again -> free padding.
// ---------------------------------------------------------------------------
__global__ void CLIR_zero_ws(float* __restrict__ p, int n) {
  int i = blockIdx.x * blockDim.x + threadIdx.x;
  if (i < n) p[i] = 0.0f;
}

// ---------------------------------------------------------------------------
// Kernel 2: streaming segment statistics (HBM-bound: 512 MB @ 23.3 TB/s).
//   Each wave processes whole rows; lane L holds columns 4L..4L+3.
//   t[row] is wave-uniform -> readfirstlane + scalar branch per class, so the
//   hot loop is 1x global_load_b128 (non-temporal) + ~9 VALU per 128 elements.
//   Per-wave partials -> LDS (ds_add_f32) -> one global atomicAdd set per WG.
// ---------------------------------------------------------------------------
__global__ __launch_bounds__(256) void CLIR_segstats(
    const float* __restrict__ x, const int* __restrict__ t,
    float* __restrict__ s1g, float* __restrict__ qg, float* __restrict__ cg,
    int n) {
  __shared__ float s1s[NUM_CLASSES * DFEAT];
  __shared__ float qs[NUM_CLASSES];
  __shared__ float cs[NUM_CLASSES];
  for (int i = threadIdx.x; i < NUM_CLASSES * DFEAT; i += blockDim.x) s1s[i] = 0.0f;
  if (threadIdx.x < NUM_CLASSES) { qs[threadIdx.x] = 0.0f; cs[threadIdx.x] = 0.0f; }
  __syncthreads();

  const int lane = threadIdx.x & 31;
  const int wave = blockIdx.x * (blockDim.x >> 5) + (threadIdx.x >> 5);
  const int nwav = gridDim.x * (blockDim.x >> 5);

  float s1[NUM_CLASSES][4];
  float q[NUM_CLASSES];
  float cnt[NUM_CLASSES];
#pragma unroll
  for (int k = 0; k < NUM_CLASSES; k++) {
    s1[k][0] = s1[k][1] = s1[k][2] = s1[k][3] = 0.0f;
    q[k] = 0.0f;
    cnt[k] = 0.0f;
  }

  for (int row = wave; row < n; row += nwav) {
    const v4f* xr = (const v4f*)(x + (size_t)row * DFEAT);
    v4f v = __builtin_nontemporal_load(&xr[lane]);           // TH_NT stream
    int c = __builtin_amdgcn_readfirstlane(t[row]);          // wave-uniform class
#pragma unroll
    for (int k = 0; k < NUM_CLASSES; k++) {
      if (c == k) {                                          // scalar branch
        s1[k][0] += v.x; s1[k][1] += v.y; s1[k][2] += v.z; s1[k][3] += v.w;
        q[k] = fmaf(v.x, v.x, q[k]);
        q[k] = fmaf(v.y, v.y, q[k]);
        q[k] = fmaf(v.z, v.z, q[k]);
        q[k] = fmaf(v.w, v.w, q[k]);
        cnt[k] += 1.0f;
      }
    }
  }

  // Wave -> LDS reduction. Lane L owns columns 4L..4L+3 (distinct addresses).
#pragma unroll
  for (int k = 0; k < NUM_CLASSES; k++) {
    const int col = lane * 4;
    atomicAdd(&s1s[k * DFEAT + col + 0], s1[k][0]);
    atomicAdd(&s1s[k * DFEAT + col + 1], s1[k][1]);
    atomicAdd(&s1s[k * DFEAT + col + 2], s1[k][2]);
    atomicAdd(&s1s[k * DFEAT + col + 3], s1[k][3]);
    float v = q[k];
#pragma unroll
    for (int off = 16; off > 0; off >>= 1) v += __shfl_down(v, off, 32);
    if (lane == 0) {
      atomicAdd(&qs[k], v);
      atomicAdd(&cs[k], cnt[k]);   // cnt is lane-uniform; lane 0's copy suffices
    }
  }
  __syncthreads();

  // Workgroup -> global: only gridDim.x updates per bin (S1 rows 0..9 only).
  for (int i = threadIdx.x; i < NUM_CLASSES * DFEAT; i += blockDim.x)
    atomicAdd(&s1g[i], s1s[i]);
  if (threadIdx.x < NUM_CLASSES) {
    atomicAdd(&qg[threadIdx.x], qs[threadIdx.x]);
    atomicAdd(&cg[threadIdx.x], cs[threadIdx.x]);
  }
}

// ---------------------------------------------------------------------------
// Kernel 3: single-wave WMMA finalize.
//   diag(S1 * S1^T) via 32 chained V_WMMA_F32_16X16X4_F32 (exact fp32 math).
//   S1 is padded to 16 zero rows -> unconditional global_load_b64 per step,
//   no EXEC masking. A-tile (16x4 of S1) and B-tile (4x16 of S1^T) have the
//   identical per-lane VGPR image (lane = idx + 16*(k/2), vgpr = k%2), so one
//   v2f load feeds both operands.
// ---------------------------------------------------------------------------
__global__ __launch_bounds__(32) void CLIR_finalize(
    const float* __restrict__ s1g, const float* __restrict__ qg,
    const float* __restrict__ cg, float* __restrict__ out) {
  const int L = threadIdx.x;        // 0..31, EXEC all ones (32-thread block)
  const int m = L & 15;             // class row (A) == column (B)
  const int koff = (L >> 4) << 1;   // lanes 0-15: K=0,1 ; lanes 16-31: K=2,3
  const float* base = s1g + m * DFEAT + koff;   // 8-byte aligned (koff even)

  v8f acc = {0.f, 0.f, 0.f, 0.f, 0.f, 0.f, 0.f, 0.f};
#pragma unroll
  for (int k4 = 0; k4 < DFEAT; k4 += 4) {
    v2f ab = *(const v2f*)(base + k4);          // global_load_b64, no guard
    acc = __builtin_amdgcn_wmma_f32_16x16x4_f32(
        /*neg_a=*/false, ab, /*neg_b=*/false, ab,
        /*c_mod=*/(short)0, acc, /*reuse_a=*/false, /*reuse_b=*/false);
  }

  // Scatter the 16x16 Gram matrix to LDS (we only need the diagonal).
  __shared__ float g[16][16];
#pragma unroll
  for (int j = 0; j < 8; j++) {
    const int M = (L < 16) ? j : (j + 8);   // C/D layout: vgpr j = row j / j+8
    g[M][m] = acc[j];
  }
  __syncthreads();

  if (L == 0) {
    float vc = 0.0f;
    for (int c = 0; c < NUM_CLASSES; c++) {
      float cn = cg[c];
      float tr = (qg[c] - g[c][c] / cn) / (cn - 1.0f);   // Q - |S1|^2/n, /(n-1)
      vc += tr;
    }
    out[0] = vc / (float)NUM_CLASSES;
  }
}

// ---------------------------------------------------------------------------
extern "C" void kernel_launch(void* const* d_in, const int* in_sizes, int n_in,
                              void* d_out, int out_size, void* d_ws, size_t ws_size,
                              hipStream_t stream) {
  const float* x = (const float*)d_in[0];
  const int* t = (const int*)d_in[1];
  const int n = in_sizes[1];                 // number of rows (t length)

  float* ws  = (float*)d_ws;
  float* s1g = ws;                           // [PAD_CLASSES * DFEAT], rows 10..15 = 0
  float* qg  = s1g + PAD_CLASSES * DFEAT;    // [NUM_CLASSES]
  float* cg  = qg + NUM_CLASSES;             // [NUM_CLASSES]
  const int nacc = PAD_CLASSES * DFEAT + 2 * NUM_CLASSES;

  CLIR_zero_ws<<<(nacc + 255) / 256, 256, 0, stream>>>(ws, nacc);
  CLIR_segstats<<<1024, 256, 0, stream>>>(x, t, s1g, qg, cg, n);
  CLIR_finalize<<<1, 32, 0, stream>>>(s1g, qg, cg, (float*)d_out);
}